// MoERouter_62380105007486
// MI455X (gfx1250) — compile-verified
//
#include <hip/hip_runtime.h>

// MoE router: scores[b,e] = silu_thresh(diff+0.5)*silu_thresh(0.5-diff), diff = opcode[b]-e
// + argmax index. B=2^21, E=64, SCALE=20.
//
// Memory-bound: 512MB score output dominates (-> non-temporal b32 stores).
// diff generation is a rank-1 update done on the matrix pipe:
//   D(16x16) = A(16x4: opcode in col 0) x B(4x16: ones in row 0) + C(= -e)
// via V_WMMA_F32_16X16X4_F32; 4 expert tiles per wave cover e=0..63 for 16 b's.
// Score math: 1 v_exp_f32 + 3 v_rcp_f32 per element (shared-exponential form).

typedef __attribute__((ext_vector_type(2))) float v2f;
typedef __attribute__((ext_vector_type(8))) float v8f;

#define NEXP 64

__global__ __launch_bounds__(256) void moe_router_wmma_kernel(
    const float* __restrict__ opcode,
    float* __restrict__ scores,
    int* __restrict__ sel,
    int batch) {
  const int lane   = threadIdx.x & 31;
  const int lane15 = lane & 15;
  const int half   = lane >> 4;                       // 0: rows 0-7, 1: rows 8-15
  const int wave   = (blockIdx.x * blockDim.x + threadIdx.x) >> 5;
  const int B0     = wave * 16;                       // 16 batch rows per wave
  if (B0 >= batch) return;                            // wave-uniform: EXEC stays all-1s

  // ---- Build WMMA operands ------------------------------------------------
  // A: 16x4 f32, A[m,0]=opcode[B0+m], A[m,1..3]=0.
  //   Layout: lanes 0-15 hold K=0 (vgpr0) / K=1 (vgpr1); lanes 16-31 K=2/K=3.
  const float op = opcode[B0 + lane15];               // halves load same 16 vals (L0 hit)
  v2f A;  A.x = half ? 0.0f : op;   A.y = 0.0f;
  // B: 4x16 f32, B[0,n]=1, rest 0 (row K striped across lanes like A).
  v2f Bm; Bm.x = half ? 0.0f : 1.0f; Bm.y = 0.0f;

  const float LOG2E  = 1.4426950408889634f;
  const float EXPM20 = 2.0611536224385578e-9f;        // e^-20
  const float SC     = 20.0f;

  float bestv[8];
  int   besti[8];
#pragma unroll
  for (int r = 0; r < 8; ++r) { bestv[r] = -1.0e30f; besti[r] = 0; }

  // scores index for (row r, tile t): (B0 + half*8 + r)*64 + t*16 + lane15
  float* rowbase = scores + (size_t)(B0 + half * 8) * NEXP + lane15;

#pragma unroll
  for (int t = 0; t < 4; ++t) {
    // C[m,n] = -(t*16 + n); in C/D layout every vgpr of a lane holds same n.
    const float e0 = (float)(t * 16 + lane15);
    v8f C;
#pragma unroll
    for (int i = 0; i < 8; ++i) C[i] = -e0;

    // D[m,n] = opcode[B0+m] - (t*16+n)   (matrix pipe)
    v8f D = __builtin_amdgcn_wmma_f32_16x16x4_f32(
        /*neg_a=*/false, A, /*neg_b=*/false, Bm,
        /*c_mod=*/(short)0, C, /*reuse_a=*/false, /*reuse_b=*/false);

#pragma unroll
    for (int r = 0; r < 8; ++r) {
      const float u  = SC * D[r];                         // 20*diff
      const float e  = __builtin_amdgcn_exp2f(-u * LOG2E); // e^{-u}; saturates safely
      const float r0 = __builtin_amdgcn_rcpf(1.0f + e);
      const float r1 = __builtin_amdgcn_rcpf(__builtin_fmaf(e, EXPM20, 1.0f));
      const float r2 = __builtin_amdgcn_rcpf(e + EXPM20);
      const float s1 = (u + SC) * r1;                     // silu(u+20)
      const float s2 = u * r0;                            // silu(u)
      const float s3 = (SC - u) * (1.0f - EXPM20 * r2);   // silu(20-u)
      const float s4 = -u * (1.0f - r0);                  // silu(-u)
      const float sc = (s1 - s2) * (s3 - s4) * 0.0025f;   // upper*lower (1/400)

      // streaming store: output (512MB) > L2 (192MB), never re-read
      __builtin_nontemporal_store(sc, rowbase + (size_t)r * NEXP + t * 16);

      const int idx = t * 16 + lane15;
      if (sc > bestv[r] || (sc == bestv[r] && idx < besti[r])) {
        bestv[r] = sc; besti[r] = idx;
      }
    }
  }

  // ---- argmax across the 16 lanes of each half-wave (first-index tiebreak) --
#pragma unroll
  for (int r = 0; r < 8; ++r) {
    float bv = bestv[r];
    int   bi = besti[r];
#pragma unroll
    for (int m = 8; m >= 1; m >>= 1) {                  // masks <16 keep halves apart
      const float ov = __shfl_xor(bv, m, 32);
      const int   oi = __shfl_xor(bi, m, 32);
      if (ov > bv || (ov == bv && oi < bi)) { bv = ov; bi = oi; }
    }
    if (lane15 == 0) sel[B0 + half * 8 + r] = bi;
  }
}

extern "C" void kernel_launch(void* const* d_in, const int* in_sizes, int n_in,
                              void* d_out, int out_size, void* d_ws, size_t ws_size,
                              hipStream_t stream) {
  const float* opcode = (const float*)d_in[0];
  const int batch = in_sizes[0];                        // 2^21 (multiple of 16)

  float* scores = (float*)d_out;                        // [batch, 64] f32
  int*   sel    = (int*)((float*)d_out + (size_t)batch * NEXP); // [batch] i32

  const int waves   = (batch + 15) / 16;                // 16 rows per wave
  const int threads = 256;                              // 8 wave32 per block
  const int blocks  = (waves * 32 + threads - 1) / threads;

  moe_router_wmma_kernel<<<blocks, threads, 0, stream>>>(opcode, scores, sel, batch);
}